// TemporalGNN_81183471829636
// MI455X (gfx1250) — compile-verified
//
#include <hip/hip_runtime.h>
#include <hip/hip_bf16.h>
#include <math.h>

// Problem constants (match reference)
#define B_   4
#define N_   30000
#define FIN_ 2
#define T_   12
#define HID_ 32
#define HOR_ 12
#define E_   240000

typedef __attribute__((ext_vector_type(2))) float v2f;
typedef __attribute__((ext_vector_type(8))) float v8f;

// D = A(16x4,f32) x B(4x16,f32) + C(16x16,f32)
__device__ __forceinline__ v8f wmma4(v2f a, v2f b, v8f c) {
  return __builtin_amdgcn_wmma_f32_16x16x4_f32(
      /*neg_a=*/false, a, /*neg_b=*/false, b,
      /*c_mod=*/(short)0, c, /*reuse_a=*/false, /*reuse_b=*/false);
}

// Branch-free activations on the HW transcendental unit (v_exp_f32 / v_rcp_f32).
// 1 - sigmoid(p) = 1/(1+exp(p))
__device__ __forceinline__ float one_minus_sigmoid(float p) {
  return __builtin_amdgcn_rcpf(1.f + __expf(p));
}
// tanh(x) = 1 - 2/(1+exp(2x)); exp overflow -> inf -> rcp -> 0 gives exact +/-1 saturation.
__device__ __forceinline__ float fast_tanh(float x) {
  return 1.f - 2.f * __builtin_amdgcn_rcpf(1.f + __expf(2.f * x));
}

// ---------------- graph preprocessing ----------------

__global__ void k_zero(float* __restrict__ p, int n) {
  int i = blockIdx.x * blockDim.x + threadIdx.x;
  if (i < n) p[i] = 0.f;
}

__global__ void k_deg(const int* __restrict__ dst, float* __restrict__ deg, int e) {
  int i = blockIdx.x * blockDim.x + threadIdx.x;
  if (i < e) atomicAdd(&deg[dst[i]], 1.0f);
}

__global__ void k_dinv(float* __restrict__ d, int n) {
  int i = blockIdx.x * blockDim.x + threadIdx.x;
  if (i < n) d[i] = rsqrtf(d[i] + 1.0f);  // +1 self loop
}

// Y[b,n,:] = dinv[n]^2 * x[b,n,:]   (self-loop term; also initializes Y)
__global__ void k_selfloop(const float* __restrict__ x, const float* __restrict__ dinv,
                           float* __restrict__ Y, int total /*B*N*/) {
  int i = blockIdx.x * blockDim.x + threadIdx.x;
  if (i >= total) return;
  int n = i % N_;
  float s = dinv[n] * dinv[n];
  const float4* xs = (const float4*)(x + (size_t)i * 24);
  float4* yd = (float4*)(Y + (size_t)i * 24);
#pragma unroll
  for (int j = 0; j < 6; ++j) {
    float4 v = xs[j];
    v.x *= s; v.y *= s; v.z *= s; v.w *= s;
    yd[j] = v;
  }
}

// Y[b,dst,:] += dinv[src]*dinv[dst] * x[b,src,:]   (all 24 = F_IN*T features at once)
__global__ void k_edges(const float* __restrict__ x, const int* __restrict__ src,
                        const int* __restrict__ dst, const float* __restrict__ dinv,
                        float* __restrict__ Y, int total /*B*E*/) {
  int i = blockIdx.x * blockDim.x + threadIdx.x;
  if (i >= total) return;
  int e = i % E_;
  int b = i / E_;
  int s = src[e], d = dst[e];
  float w = dinv[s] * dinv[d];
  const float* xs = x + ((size_t)b * N_ + s) * 24;
  float* yd = Y + ((size_t)b * N_ + d) * 24;
#pragma unroll
  for (int j = 0; j < 24; ++j) atomicAdd(yd + j, w * xs[j]);
}

// ---------------- parameter fusion ----------------
// P[0..63]   Wz = conv_z_w @ lin_z_w[0:32]   (f-major: P[f*32+h])
// P[64..127] Wh = conv_h_w @ lin_h_w[0:32]
// P[128..159] bz = lin_z_b + conv_z_b @ lin_z_w[0:32]
// P[160..191] bh = lin_h_b + conv_h_b @ lin_h_w[0:32]
// P[192..203] probs = softmax(attention)
__global__ void k_prep(const float* __restrict__ att,
                       const float* __restrict__ czw, const float* __restrict__ czb,
                       const float* __restrict__ lzw, const float* __restrict__ lzb,
                       const float* __restrict__ chw, const float* __restrict__ chb,
                       const float* __restrict__ lhw, const float* __restrict__ lhb,
                       float* __restrict__ P) {
  int tid = threadIdx.x;  // 64 threads
  int f = tid >> 5, h = tid & 31;
  float wz = 0.f, wh = 0.f;
  for (int k = 0; k < HID_; ++k) {
    wz += czw[f * HID_ + k] * lzw[k * HID_ + h];
    wh += chw[f * HID_ + k] * lhw[k * HID_ + h];
  }
  P[tid] = wz;
  P[64 + tid] = wh;
  if (tid < HID_) {
    float bz = lzb[h], bh = lhb[h];
    for (int k = 0; k < HID_; ++k) {
      bz += czb[k] * lzw[k * HID_ + h];
      bh += chb[k] * lhw[k * HID_ + h];
    }
    P[128 + h] = bz;
    P[160 + h] = bh;
  }
  if (tid < T_) {
    float m = att[0];
    for (int t = 1; t < T_; ++t) m = fmaxf(m, att[t]);
    float s = 0.f;
    for (int t = 0; t < T_; ++t) s += __expf(att[t] - m);
    P[192 + tid] = __expf(att[tid] - m) / s;
  }
}

// ---------------- fused GRU-gate + attention + head (WMMA) ----------------
// One wave per 16 nodes. 128 threads = 4 waves/block; 7500 tiles = 1875 blocks (all active).
__global__ void __launch_bounds__(128) k_fused(
    const float* __restrict__ Y, const float* __restrict__ P,
    const float* __restrict__ head_w, const float* __restrict__ head_b,
    float* __restrict__ out, int totalRows) {
  __shared__ float lds[4][16 * 33];  // per-wave 16x32 transpose tile (+pad)
  const int wave = threadIdx.x >> 5;
  const int lane = threadIdx.x & 31;
  const int tile = blockIdx.x * 4 + wave;
  const int rowBase = tile * 16;
  const bool lo = lane < 16;
  const int ln = lane & 15;
  if (rowBase >= totalRows) return;  // never taken (exact grid), kept for safety

  // A-fragment source data: lane m<16 carries node rowBase+m (K=0,1); lanes 16..31 are K-pad.
  float y0[T_], y1[T_];
  {
    const float* yp = Y + (size_t)(rowBase + ln) * 24;
#pragma unroll
    for (int t = 0; t < T_; ++t) {
      y0[t] = lo ? yp[t] : 0.f;        // f=0
      y1[t] = lo ? yp[T_ + t] : 0.f;   // f=1
    }
  }

  // B fragments (4x16, K rows 2..3 zero-padded), time-invariant.
  v2f Bz0, Bz1, Bh0, Bh1;
  Bz0[0] = lo ? P[ln] : 0.f;        Bz0[1] = lo ? P[32 + ln] : 0.f;
  Bz1[0] = lo ? P[16 + ln] : 0.f;   Bz1[1] = lo ? P[48 + ln] : 0.f;
  Bh0[0] = lo ? P[64 + ln] : 0.f;   Bh0[1] = lo ? P[96 + ln] : 0.f;
  Bh1[0] = lo ? P[80 + ln] : 0.f;   Bh1[1] = lo ? P[112 + ln] : 0.f;

  const float bz_lo = P[128 + ln], bz_hi = P[144 + ln];
  const float bh_lo = P[160 + ln], bh_hi = P[176 + ln];

  float pv[T_];
#pragma unroll
  for (int t = 0; t < T_; ++t) pv[t] = P[192 + t];

  const v8f czero = {0.f, 0.f, 0.f, 0.f, 0.f, 0.f, 0.f, 0.f};
  v8f acc0 = czero, acc1 = czero;

#pragma unroll
  for (int t = 0; t < T_; ++t) {
    v2f A;
    A[0] = y0[t];  // K=0 (lanes 0-15) / K=2 pad=0 (lanes 16-31)
    A[1] = y1[t];  // K=1 / K=3 pad
    v8f cz0 = wmma4(A, Bz0, czero);
    v8f cz1 = wmma4(A, Bz1, czero);
    v8f ch0 = wmma4(A, Bh0, czero);
    v8f ch1 = wmma4(A, Bh1, czero);
    const float pt = pv[t];
#pragma unroll
    for (int i = 0; i < 8; ++i) {
      // Hn = (1-Z)*Ht  (R-gate is dead: H0==0)
      float omz0 = one_minus_sigmoid(cz0[i] + bz_lo);
      float th0  = fast_tanh(ch0[i] + bh_lo);
      acc0[i] += pt * omz0 * th0;
      float omz1 = one_minus_sigmoid(cz1[i] + bz_hi);
      float th1  = fast_tanh(ch1[i] + bh_hi);
      acc1[i] += pt * omz1 * th1;
    }
  }

  // relu(acc): C layout -> A layout transpose through LDS
  const int rbase = (lane >> 4) * 8;  // C row M = k + 8*(lane/16), col = lane&15
#pragma unroll
  for (int k = 0; k < 8; ++k) {
    lds[wave][(rbase + k) * 33 + ln] = fmaxf(acc0[k], 0.f);
    lds[wave][(rbase + k) * 33 + 16 + ln] = fmaxf(acc1[k], 0.f);
  }
  __syncthreads();  // all waves active; orders LDS store->load

  // head: (16x32) @ (32x12 padded to 16) via 8 K-chunks of WMMA 16x16x4
  const int koff = (lane >> 4) * 2;  // A/B VGPR0 K-subindex for this half-wave
  v8f hacc = czero;
#pragma unroll
  for (int c = 0; c < 8; ++c) {
    const int k0 = 4 * c + koff;
    v2f A, Bf;
    A[0] = lds[wave][ln * 33 + k0];
    A[1] = lds[wave][ln * 33 + k0 + 1];
    Bf[0] = (ln < HOR_) ? head_w[k0 * HOR_ + ln] : 0.f;
    Bf[1] = (ln < HOR_) ? head_w[(k0 + 1) * HOR_ + ln] : 0.f;
    hacc = wmma4(A, Bf, hacc);
  }

  const float hbv = (ln < HOR_) ? head_b[ln] : 0.f;
#pragma unroll
  for (int k = 0; k < 8; ++k) {
    if (ln < HOR_) {
      int row = rowBase + rbase + k;
      out[(size_t)row * HOR_ + ln] = hacc[k] + hbv;
    }
  }
}

// ---------------- launcher ----------------

extern "C" void kernel_launch(void* const* d_in, const int* in_sizes, int n_in,
                              void* d_out, int out_size, void* d_ws, size_t ws_size,
                              hipStream_t stream) {
  (void)in_sizes; (void)n_in; (void)out_size; (void)ws_size;
  const float* x   = (const float*)d_in[0];   // (B,N,F_IN,T): node-contiguous 24 floats
  const int*   ei  = (const int*)d_in[1];     // (2,E)
  const float* att = (const float*)d_in[2];
  const float* czw = (const float*)d_in[3];
  const float* czb = (const float*)d_in[4];
  const float* lzw = (const float*)d_in[5];
  const float* lzb = (const float*)d_in[6];
  // d_in[7..10] = r-gate params: mathematically dead (H0==0 => H0*R==0)
  const float* chw = (const float*)d_in[11];
  const float* chb = (const float*)d_in[12];
  const float* lhw = (const float*)d_in[13];
  const float* lhb = (const float*)d_in[14];
  const float* hw  = (const float*)d_in[15];
  const float* hb  = (const float*)d_in[16];
  float* out = (float*)d_out;

  // workspace: dinv (N) | Y (B*N*24) | P (256)   ~= 11.7 MB
  float* ws   = (float*)d_ws;
  float* dinv = ws;
  float* Y    = ws + 30016;
  float* P    = Y + (size_t)B_ * N_ * 24;

  const int* src = ei;
  const int* dst = ei + E_;

  const int TB = 256;
  k_zero<<<(N_ + TB - 1) / TB, TB, 0, stream>>>(dinv, N_);
  k_deg<<<(E_ + TB - 1) / TB, TB, 0, stream>>>(dst, dinv, E_);
  k_dinv<<<(N_ + TB - 1) / TB, TB, 0, stream>>>(dinv, N_);
  k_selfloop<<<(B_ * N_ + TB - 1) / TB, TB, 0, stream>>>(x, dinv, Y, B_ * N_);
  k_edges<<<(B_ * E_ + TB - 1) / TB, TB, 0, stream>>>(x, src, dst, dinv, Y, B_ * E_);
  k_prep<<<1, 64, 0, stream>>>(att, czw, czb, lzw, lzb, chw, chb, lhw, lhb, P);

  const int totalRows = B_ * N_;          // 120000
  const int tiles = totalRows / 16;       // 7500
  k_fused<<<tiles / 4, 128, 0, stream>>>(Y, P, hw, hb, out, totalRows);
}